// CapsuleLayer_6339371729212
// MI455X (gfx1250) — compile-verified
//
#include <hip/hip_runtime.h>
#include <math.h>

// ---------------------------------------------------------------------------
// CapsNet dynamic routing on gfx1250 (MI455X), fp32 WMMA path.
//
// Shapes: x[B=128][I=1152][L=8], W[J=32][I=1152][K=16][L=8]
// Never materialize u_hat (302MB). Both big einsums are per-j GEMMs against
// W[j] (which is L2-resident, 18.9MB total):
//   s[b,j,k]   = sum_{i,l} W[j,i,k,l] * (c[b,j,i]*x[b,i,l])   (M=128,N=16,K=9216)
//   db[b,j,i]  = sum_l x[b,i,l] * sum_k W[j,i,k,l]*v[b,j,k]   (M=128,N=9216,K=16)
// Mapped onto V_WMMA_F32_16X16X4_F32 (D = A(16x4) * B(4x16) + C).
// Scratch (d_ws): bb[128][32][1152] logits + c[128][32][1152] softmax.
// UNIFORM (round 0, c = 1/32) is a template parameter so the inner WMMA loop
// is branch-free in both variants.
// ---------------------------------------------------------------------------

typedef float v2f __attribute__((ext_vector_type(2)));
typedef float v8f __attribute__((ext_vector_type(8)));

#define BSZ 128
#define JC  32
#define IC  1152
#define KD  16
#define LD  8
// x stride per b = IC*LD = 9216 ; W stride per j = IC*KD*LD = 147456

__global__ void caps_zero_kernel(float4* __restrict__ p, int n4) {
  int t = blockIdx.x * blockDim.x + threadIdx.x;
  if (t < n4) p[t] = make_float4(0.f, 0.f, 0.f, 0.f);
}

// c[b,j,i] = softmax over j of bb[b,j,i]; one thread per (b,i)
__global__ void caps_softmax_kernel(const float* __restrict__ bb,
                                    float* __restrict__ c) {
  int t = blockIdx.x * blockDim.x + threadIdx.x;   // 0 .. B*IC-1
  int b = t / IC;
  int i = t - b * IC;
  const float* p = bb + (size_t)b * JC * IC + i;
  float vals[JC];
  float m = -1e30f;
#pragma unroll
  for (int j = 0; j < JC; ++j) { vals[j] = p[(size_t)j * IC]; m = fmaxf(m, vals[j]); }
  float s = 0.f;
#pragma unroll
  for (int j = 0; j < JC; ++j) { vals[j] = __expf(vals[j] - m); s += vals[j]; }
  float inv = 1.f / s;
  float* q = c + (size_t)b * JC * IC + i;
#pragma unroll
  for (int j = 0; j < JC; ++j) q[(size_t)j * IC] = vals[j] * inv;
}

// s[b,j,:] GEMM + squash -> vout[b,j,:].  grid = (j=32, btile=8), 256 thr.
template <bool UNIFORM>
__global__ __launch_bounds__(256)
void caps_s_kernel(const float* __restrict__ x, const float* __restrict__ W,
                   const float* __restrict__ c, float* __restrict__ vout) {
  const int j    = blockIdx.x;
  const int b0   = blockIdx.y * 16;
  const int lane = threadIdx.x & 31;
  const int w    = threadIdx.x >> 5;     // 8 waves split the i (K) range
  const int mrow = lane & 15;            // M row (batch within tile) / N col for B frag
  const int lhi  = lane >> 4;
  const int b    = b0 + mrow;

  const float* Wj = W + (size_t)j * (IC * KD * LD);
  const float* xb = x + (size_t)b * (IC * LD);
  const float* cb = c + ((size_t)b * JC + j) * IC;

  v8f acc0 = {}; v8f acc1 = {};
  const int i_beg = w * (IC / 8);
  const int i_end = i_beg + (IC / 8);
  for (int i = i_beg; i < i_end; i += 2) {
    // ---- i (even) ----
    {
      float cv = UNIFORM ? (1.0f / 32.0f) : cb[i];
      const float* xi = xb + (size_t)i * LD + 2 * lhi;      // A[m, l]
      v2f a0 = { cv * xi[0], cv * xi[1] };                  // l = 2*lhi, 2*lhi+1
      v2f a1 = { cv * xi[4], cv * xi[5] };                  // l = 4+2*lhi, ...
      const float* Wi = Wj + (size_t)i * (KD * LD) + mrow * LD + 2 * lhi; // B[l, k=n]
      v2f f0 = { Wi[0], Wi[1] };
      v2f f1 = { Wi[4], Wi[5] };
      acc0 = __builtin_amdgcn_wmma_f32_16x16x4_f32(false, a0, false, f0, (short)0, acc0, false, false);
      acc0 = __builtin_amdgcn_wmma_f32_16x16x4_f32(false, a1, false, f1, (short)0, acc0, false, false);
    }
    // ---- i+1 (odd), independent accumulator to hide WMMA latency ----
    {
      int i1 = i + 1;
      float cv = UNIFORM ? (1.0f / 32.0f) : cb[i1];
      const float* xi = xb + (size_t)i1 * LD + 2 * lhi;
      v2f a0 = { cv * xi[0], cv * xi[1] };
      v2f a1 = { cv * xi[4], cv * xi[5] };
      const float* Wi = Wj + (size_t)i1 * (KD * LD) + mrow * LD + 2 * lhi;
      v2f f0 = { Wi[0], Wi[1] };
      v2f f1 = { Wi[4], Wi[5] };
      acc1 = __builtin_amdgcn_wmma_f32_16x16x4_f32(false, a0, false, f0, (short)0, acc1, false, false);
      acc1 = __builtin_amdgcn_wmma_f32_16x16x4_f32(false, a1, false, f1, (short)0, acc1, false, false);
    }
  }

  // ---- cross-wave reduction of the 16x16 accumulator tile via LDS ----
  __shared__ float red[8][256];
#pragma unroll
  for (int r = 0; r < 8; ++r) red[w][lane + 32 * r] = acc0[r] + acc1[r];
  __syncthreads();

  const int t = threadIdx.x;
  float sum = 0.f;
#pragma unroll
  for (int wv = 0; wv < 8; ++wv) sum += red[wv][t];

  // value at thread t corresponds to C/D element: m = r + 8*lhi', n = lane'&15
  __shared__ float sT[16][17];
  {
    int lp = t & 31;
    int m  = (t >> 5) + 8 * (lp >> 4);
    int n  = t & 15;
    sT[m][n] = sum;
  }
  __syncthreads();

  __shared__ float scl[16];
  if (t < 16) {
    float s2 = 0.f;
#pragma unroll
    for (int k = 0; k < KD; ++k) { float q = sT[t][k]; s2 += q * q; }
    scl[t] = s2 / (1.0f + s2) / sqrtf(s2 + 1e-7f);   // squash scale
  }
  __syncthreads();

  int mm = t >> 4, kk = t & 15;
  vout[(((size_t)(b0 + mm)) * JC + j) * KD + kk] = scl[mm] * sT[mm][kk];
}

// bb[b,j,i] += sum_l x[b,i,l] * (W[j]^T v)[b,i,l].  grid = (j=32, btile=8).
__global__ __launch_bounds__(256)
void caps_agree_kernel(const float* __restrict__ x, const float* __restrict__ W,
                       const float* __restrict__ v, float* __restrict__ bb) {
  const int j    = blockIdx.x;
  const int b0   = blockIdx.y * 16;
  const int lane = threadIdx.x & 31;
  const int w    = threadIdx.x >> 5;
  const int lhi  = lane >> 4;
  const int mrow = lane & 15;          // N column index n for B/D frags
  const int ii   = mrow >> 3;          // which of the 2 i's in this column block
  const int ll   = mrow & 7;           // l within capsule row

  // A = v tile (16x16), fixed for whole block: A[m, kk] = v[b0+m, j, kk]
  v2f a[4];
  {
    const float* vp = v + (((size_t)(b0 + mrow)) * JC + j) * KD;
#pragma unroll
    for (int tk = 0; tk < 4; ++tk) {
      a[tk].x = vp[4 * tk + 2 * lhi];
      a[tk].y = vp[4 * tk + 2 * lhi + 1];
    }
  }
  const float* Wj = W + (size_t)j * (IC * KD * LD);

  const int np = (IC / 2) / 8;          // 72 i-pairs per wave
  for (int p = 0; p < np; ++p) {
    const int i0 = 2 * (w * np + p);
    const int iW = i0 + ii;
    const float* Wi = Wj + (size_t)iW * (KD * LD) + ll;
    v8f acc = {};
#pragma unroll
    for (int tk = 0; tk < 4; ++tk) {    // K = 16 over capsule dim k
      int kk = 4 * tk + 2 * lhi;
      v2f f = { Wi[(size_t)kk * LD], Wi[(size_t)(kk + 1) * LD] };  // B[kk, n] = W[j,iW,kk,ll]
      acc = __builtin_amdgcn_wmma_f32_16x16x4_f32(false, a[tk], false, f, (short)0, acc, false, false);
    }
    // D[r] = g[b0 + r + 8*lhi, j, iW, ll]; multiply by x and reduce over l lanes
    float pr[8];
#pragma unroll
    for (int r = 0; r < 8; ++r) {
      int bidx = b0 + r + 8 * lhi;
      pr[r] = acc[r] * x[((size_t)bidx) * (IC * LD) + (size_t)iW * LD + ll];
    }
#pragma unroll
    for (int r = 0; r < 8; ++r) {       // butterfly over the 8 l-lanes
      pr[r] += __shfl_xor(pr[r], 1, 32);
      pr[r] += __shfl_xor(pr[r], 2, 32);
      pr[r] += __shfl_xor(pr[r], 4, 32);
    }
    if (ll == 0) {                      // one writer per (i, m-half)
#pragma unroll
      for (int r = 0; r < 8; ++r) {
        size_t idx = (((size_t)(b0 + r + 8 * lhi)) * JC + j) * IC + iW;
        bb[idx] += pr[r];
      }
    }
  }
}

extern "C" void kernel_launch(void* const* d_in, const int* in_sizes, int n_in,
                              void* d_out, int out_size, void* d_ws, size_t ws_size,
                              hipStream_t stream) {
  const float* x = (const float*)d_in[0];   // inputs [128,1152,8]
  const float* W = (const float*)d_in[1];   // W      [32,1152,16,8]
  float* vout = (float*)d_out;              // v      [128,32,16]

  float* bb = (float*)d_ws;                 // logits [128,32,1152]
  float* cc = bb + (size_t)BSZ * JC * IC;   // softmax coeffs, same size

  // bb must start at zero every call (deterministic replay)
  const int n4 = (BSZ * JC * IC) / 4;
  caps_zero_kernel<<<(n4 + 255) / 256, 256, 0, stream>>>((float4*)bb, n4);

  dim3 gG(JC, BSZ / 16);                    // 32 x 8 workgroups, 256 threads

  // round 0: c is uniform 1/32 (branch-free template instantiation)
  caps_s_kernel<true><<<gG, 256, 0, stream>>>(x, W, cc, vout);
  caps_agree_kernel<<<gG, 256, 0, stream>>>(x, W, vout, bb);
  // round 1
  caps_softmax_kernel<<<(BSZ * IC) / 256, 256, 0, stream>>>(bb, cc);
  caps_s_kernel<false><<<gG, 256, 0, stream>>>(x, W, cc, vout);
  caps_agree_kernel<<<gG, 256, 0, stream>>>(x, W, vout, bb);
  // round 2 (final v -> d_out)
  caps_softmax_kernel<<<(BSZ * IC) / 256, 256, 0, stream>>>(bb, cc);
  caps_s_kernel<false><<<gG, 256, 0, stream>>>(x, W, cc, vout);
}